// SpatialCoherence_53480932770122
// MI455X (gfx1250) — compile-verified
//
#include <hip/hip_runtime.h>
#include <hip/hip_bf16.h>
#include <cstdint>
#include <cstddef>

typedef __attribute__((ext_vector_type(2))) float v2f;
typedef __attribute__((ext_vector_type(8))) float v8f;

namespace {
constexpr int kH = 1024;
constexpr int kW = 1024;
constexpr int kPix = kH * kW;
constexpr int kBig = kPix + 1;          // sentinel label (matches reference BIG)
constexpr int kStatW = 320;             // 32-bit words of per-image stats (4 + 256 hist, padded)
constexpr int kBins = 256;
constexpr int kThreads = 256;
constexpr int kBlocksPerImg = kPix / kThreads;  // 4096
}

// Order-preserving float <-> uint encode for integer atomicMin/Max.
__device__ __forceinline__ unsigned fenc(float f) {
  unsigned u = __float_as_uint(f);
  return (u & 0x80000000u) ? ~u : (u | 0x80000000u);
}
__device__ __forceinline__ float fdec(unsigned u) {
  unsigned v = (u & 0x80000000u) ? (u & 0x7fffffffu) : ~u;
  return __uint_as_float(v);
}

// Per-image stats layout (32-bit words):
// [0]=min(enc) [1]=max(enc) [2]=thr(float bits) [3]=maxlab(int) [4..259]=hist

extern "C" __global__ void sc_init_stats(unsigned* stats, int nimg) {
  int i = blockIdx.x * blockDim.x + threadIdx.x;
  int tot = nimg * kStatW;
  if (i >= tot) return;
  int w = i % kStatW;
  unsigned v = 0u;
  if (w == 0) v = 0xFFFFFFFFu;   // min sentinel (encoded +max)
  else if (w == 3) v = 0xFFFFFFFFu;  // maxlab = -1
  stats[i] = v;
}

extern "C" __global__ void sc_minmax(const float* __restrict__ s, unsigned* stats) {
  int img = blockIdx.y;
  const float* p = s + (size_t)img * kPix;
  float mn = __uint_as_float(0x7F800000u);
  float mx = __uint_as_float(0xFF800000u);
  for (int i = blockIdx.x * blockDim.x + threadIdx.x; i < kPix;
       i += gridDim.x * blockDim.x) {
    float v = p[i];
    mn = fminf(mn, v);
    mx = fmaxf(mx, v);
  }
  __shared__ float smn[kThreads];
  __shared__ float smx[kThreads];
  smn[threadIdx.x] = mn;
  smx[threadIdx.x] = mx;
  __syncthreads();
  for (int st = kThreads / 2; st > 0; st >>= 1) {
    if (threadIdx.x < st) {
      smn[threadIdx.x] = fminf(smn[threadIdx.x], smn[threadIdx.x + st]);
      smx[threadIdx.x] = fmaxf(smx[threadIdx.x], smx[threadIdx.x + st]);
    }
    __syncthreads();
  }
  if (threadIdx.x == 0) {
    atomicMin(&stats[(size_t)img * kStatW + 0], fenc(smn[0]));
    atomicMax(&stats[(size_t)img * kStatW + 1], fenc(smx[0]));
  }
}

extern "C" __global__ void sc_hist(const float* __restrict__ s, unsigned* stats) {
  int img = blockIdx.y;
  unsigned* st = stats + (size_t)img * kStatW;
  float mn = fdec(st[0]);
  float mx = fdec(st[1]);
  float width = (mx - mn) / 256.0f;
  float safe = (width > 0.0f) ? width : 1.0f;
  __shared__ unsigned h[kBins];
  h[threadIdx.x] = 0u;
  __syncthreads();
  const float* p = s + (size_t)img * kPix;
  for (int i = blockIdx.x * blockDim.x + threadIdx.x; i < kPix;
       i += gridDim.x * blockDim.x) {
    int b = (int)floorf((p[i] - mn) / safe);
    b = b < 0 ? 0 : (b > 255 ? 255 : b);
    atomicAdd(&h[b], 1u);
  }
  __syncthreads();
  atomicAdd(&st[4 + threadIdx.x], h[threadIdx.x]);
}

// One thread per image: replicate the reference cumsum formulas exactly.
extern "C" __global__ void sc_otsu(unsigned* stats, int nimg) {
  int img = threadIdx.x;
  if (img >= nimg) return;
  unsigned* st = stats + (size_t)img * kStatW;
  float mn = fdec(st[0]);
  float mx = fdec(st[1]);
  float width = (mx - mn) / 256.0f;
  float thr;
  if (!(width > 0.0f)) {
    thr = mn;  // constant image: thr = first value = mn; mask all false
  } else {
    float revw[257], revcc[257];
    revw[256] = 0.0f;
    revcc[256] = 0.0f;
    for (int i = 255; i >= 0; --i) {
      float c = (float)st[4 + i];
      float cen = mn + ((float)i + 0.5f) * width;
      revw[i] = revw[i + 1] + c;
      revcc[i] = revcc[i + 1] + c * cen;
    }
    float w1 = 0.0f, cc1 = 0.0f, best = 0.0f;
    int bi = 0;
    bool found = false;
    for (int i = 0; i < 255; ++i) {
      float c = (float)st[4 + i];
      float cen = mn + ((float)i + 0.5f) * width;
      w1 += c;
      cc1 += c * cen;
      float w2 = revw[i + 1];
      float m1 = cc1 / w1;
      float m2 = revcc[i + 1] / w2;
      float dm = m1 - m2;
      float var = w1 * w2 * dm * dm;
      if (!found || var > best) { best = var; bi = i; found = true; }
    }
    thr = mn + ((float)bi + 0.5f) * width;
  }
  ((float*)st)[2] = thr;
}

// Fused mask (> thr), cross-shaped erosion (border_value=0), label init.
extern "C" __global__ void sc_label_init(const float* __restrict__ s,
                                         const unsigned* __restrict__ stats,
                                         int* __restrict__ labels) {
  int img = blockIdx.y;
  const float* p = s + (size_t)img * kPix;
  float thr = __uint_as_float(stats[(size_t)img * kStatW + 2]);
  int i = blockIdx.x * blockDim.x + threadIdx.x;
  int y = i / kW, x = i % kW;
  bool er = (p[i] > thr);
  er = er && (y > 0) && (p[i - kW] > thr);
  er = er && (y < kH - 1) && (p[i + kW] > thr);
  er = er && (x > 0) && (p[i - 1] > thr);
  er = er && (x < kW - 1) && (p[i + 1] > thr);
  labels[(size_t)img * kPix + i] = er ? i : kBig;
}

__device__ __forceinline__ int sc_find(int* L, int x) {
  int p = __hip_atomic_load(&L[x], __ATOMIC_RELAXED, __HIP_MEMORY_SCOPE_AGENT);
  while (p != x) {
    x = p;
    p = __hip_atomic_load(&L[x], __ATOMIC_RELAXED, __HIP_MEMORY_SCOPE_AGENT);
  }
  return x;
}

__device__ __forceinline__ void sc_union(int* L, int a, int b) {
  a = sc_find(L, a);
  b = sc_find(L, b);
  while (a != b) {
    if (a < b) { int t = a; a = b; b = t; }  // a > b
    int old = atomicMin(&L[a], b);
    if (old == a) break;
    a = old;
  }
}

// 8-connectivity: each pixel unions with E, SW, S, SE (covers all pairs).
extern "C" __global__ void sc_merge(int* labels) {
  int img = blockIdx.y;
  int* L = labels + (size_t)img * kPix;
  int i = blockIdx.x * blockDim.x + threadIdx.x;
  if (L[i] == kBig) return;
  int y = i / kW, x = i % kW;
  bool e = (x < kW - 1), so = (y < kH - 1), w = (x > 0);
  if (e && L[i + 1] != kBig) sc_union(L, i, i + 1);
  if (so) {
    if (w && L[i + kW - 1] != kBig) sc_union(L, i, i + kW - 1);
    if (L[i + kW] != kBig) sc_union(L, i, i + kW);
    if (e && L[i + kW + 1] != kBig) sc_union(L, i, i + kW + 1);
  }
}

// Flatten to root (= min raster index of component) and compute maxlab.
extern "C" __global__ void sc_flatten(int* labels, unsigned* stats) {
  int img = blockIdx.y;
  int* L = labels + (size_t)img * kPix;
  int i = blockIdx.x * blockDim.x + threadIdx.x;
  int v = L[i];
  int r = -1;
  if (v != kBig) {
    r = sc_find(L, i);
    L[i] = r;
  }
  __shared__ int smax[kThreads];
  smax[threadIdx.x] = r;
  __syncthreads();
  for (int st = kThreads / 2; st > 0; st >>= 1) {
    if (threadIdx.x < st)
      smax[threadIdx.x] = max(smax[threadIdx.x], smax[threadIdx.x + st]);
    __syncthreads();
  }
  if (threadIdx.x == 0 && smax[0] >= 0)
    atomicMax((int*)&stats[(size_t)img * kStatW + 3], smax[0]);
}

// Stencil contributions + wave reduction via V_WMMA_F32_16X16X4_F32.
extern "C" __global__ void sc_sum(const float* __restrict__ s,
                                  const int* __restrict__ labels,
                                  const unsigned* __restrict__ stats,
                                  float* __restrict__ partial) {
  int img = blockIdx.y;
  const float* p = s + (size_t)img * kPix;
  const int* L = labels + (size_t)img * kPix;
  int maxlab = (int)stats[(size_t)img * kStatW + 3];
  int i = blockIdx.x * blockDim.x + threadIdx.x;
  int y = i / kW, x = i % kW;
  int lp = L[i];
  float sp = p[i];
  bool vp = (lp != kBig) && (lp != maxlab);
  float acc = 0.0f;
  const int dy[8] = {-1, -1, -1, 0, 0, 1, 1, 1};
  const int dx[8] = {-1, 0, 1, -1, 1, -1, 0, 1};
#pragma unroll
  for (int k = 0; k < 8; ++k) {
    int ny = y + dy[k], nx = x + dx[k];
    if ((unsigned)ny < (unsigned)kH && (unsigned)nx < (unsigned)kW) {
      int q = ny * kW + nx;
      int lq = L[q];
      float sq = p[q];
      bool vq = (lq != kBig) && (lq != maxlab);
      if (lp == lq) {
        if (vp) { float d = sp - sq; acc += d * d; }
      } else {
        if (vp) acc += sp * sp;
        if (vq) acc += sq * sq;
      }
    }
  }
  // Wave32 reduction: D = A x ones(4x16) + 0. Column 0 of D (lanes 0 and 16,
  // 8 VGPRs each) holds all 16 row-sums; their total equals the sum of every
  // A element = sum of acc over the 32 lanes, independent of A's lane layout.
  v2f a; a.x = acc; a.y = 0.0f;
  v2f b; b.x = 1.0f; b.y = 1.0f;
  v8f c = {};
  v8f d = __builtin_amdgcn_wmma_f32_16x16x4_f32(false, a, false, b, (short)0, c,
                                                false, false);
  __shared__ float sred[2 * (kThreads / 32)];
  int lane = threadIdx.x & 31;
  int wv = threadIdx.x >> 5;
  if ((lane & 15) == 0) {
    float wsum = 0.0f;
#pragma unroll
    for (int j = 0; j < 8; ++j) wsum += d[j];
    sred[wv * 2 + (lane >> 4)] = wsum;
  }
  __syncthreads();
  if (threadIdx.x == 0) {
    float t = 0.0f;
    for (int j = 0; j < 2 * (kThreads / 32); ++j) t += sred[j];
    partial[(size_t)img * gridDim.x + blockIdx.x] = t;
  }
}

// Deterministic final reduction (fixed order, single block).
extern "C" __global__ void sc_final(const float* __restrict__ partial, int n,
                                    float* __restrict__ out) {
  __shared__ float sm[256];
  float a = 0.0f;
  for (int i = threadIdx.x; i < n; i += 256) a += partial[i];
  sm[threadIdx.x] = a;
  __syncthreads();
  for (int st = 128; st > 0; st >>= 1) {
    if (threadIdx.x < st) sm[threadIdx.x] += sm[threadIdx.x + st];
    __syncthreads();
  }
  if (threadIdx.x == 0) out[0] = sm[0];
}

extern "C" void kernel_launch(void* const* d_in, const int* in_sizes, int n_in,
                              void* d_out, int out_size, void* d_ws, size_t ws_size,
                              hipStream_t stream) {
  (void)n_in; (void)out_size; (void)ws_size;
  const float* sal = (const float*)d_in[0];
  float* out = (float*)d_out;
  int nimg = in_sizes[0] / kPix;  // 2*4 = 8 images

  // Workspace layout: [stats: nimg*kStatW words][labels: nimg*kPix ints][partials]
  unsigned* stats = (unsigned*)d_ws;
  int* labels = (int*)((char*)d_ws + 16384);
  float* partial =
      (float*)((char*)d_ws + 16384 + (size_t)nimg * kPix * sizeof(int));

  sc_init_stats<<<(nimg * kStatW + 255) / 256, 256, 0, stream>>>(stats, nimg);
  sc_minmax<<<dim3(64, nimg), kThreads, 0, stream>>>(sal, stats);
  sc_hist<<<dim3(64, nimg), kThreads, 0, stream>>>(sal, stats);
  sc_otsu<<<1, 32, 0, stream>>>(stats, nimg);
  sc_label_init<<<dim3(kBlocksPerImg, nimg), kThreads, 0, stream>>>(sal, stats,
                                                                    labels);
  sc_merge<<<dim3(kBlocksPerImg, nimg), kThreads, 0, stream>>>(labels);
  sc_flatten<<<dim3(kBlocksPerImg, nimg), kThreads, 0, stream>>>(labels, stats);
  sc_sum<<<dim3(kBlocksPerImg, nimg), kThreads, 0, stream>>>(sal, labels, stats,
                                                             partial);
  sc_final<<<1, 256, 0, stream>>>(partial, nimg * kBlocksPerImg, out);
}